// Voxelization_2576980377720
// MI455X (gfx1250) — compile-verified
//
#include <hip/hip_runtime.h>

// Problem constants (from reference)
#define X_DIM 38
#define Y_DIM 24
#define Z_DIM 24
#define RVOX  (X_DIM * Y_DIM * Z_DIM)   // 21888, divisible by 16
#define BATCH 16
#define CHANS 64
#define NPTS  16384

typedef __attribute__((ext_vector_type(2))) float v2f;
typedef __attribute__((ext_vector_type(8))) float v8f;

// ---------------------------------------------------------------------------
// Phase 0: zero-init (float4 stores; all sizes are multiples of 4 floats)
// ---------------------------------------------------------------------------
__global__ void zero_f4(float4* __restrict__ p, int n4) {
    int i = blockIdx.x * blockDim.x + threadIdx.x;
    if (i < n4) p[i] = make_float4(0.f, 0.f, 0.f, 0.f);
}

// ---------------------------------------------------------------------------
// Phase 1: per-point flat voxel id (matches reference f32 math) + counts
// ---------------------------------------------------------------------------
__global__ void flat_count(const float* __restrict__ coords,
                           int* __restrict__ flats,
                           float* __restrict__ counts) {
    int t = blockIdx.x * blockDim.x + threadIdx.x;   // 0 .. B*N-1
    int b = t >> 14;                                  // N = 16384
    const float* c = coords + (size_t)t * 3;
    float x = c[0], y = c[1], z = c[2];
    // reference: floor(coords / vs) - MIN_VOXEL_COORD, all in f32
    float fx = floorf(x / 0.3f) + 19.0f;
    float fy = floorf(y / 0.3f) + 12.0f;
    float fz = floorf(z / 0.2f) + 12.0f;
    int ix = (int)fx, iy = (int)fy, iz = (int)fz;
    bool valid = (ix >= 0) & (ix < X_DIM) & (iy >= 0) & (iy < Y_DIM) &
                 (iz >= 0) & (iz < Z_DIM);
    int flat = valid ? (ix * (Y_DIM * Z_DIM) + iy * Z_DIM + iz) : RVOX;
    flats[t] = flat;
    if (valid) atomicAdd(&counts[b * RVOX + flat], 1.0f);
}

// ---------------------------------------------------------------------------
// Phase 2: scatter-add features into final [B, C, R] layout.
// Wave (32 lanes) = 32 consecutive points, one channel per iteration:
// feature reads are 128B-coalesced; atomics land in L2 (90 MB output < 192 MB L2).
// ---------------------------------------------------------------------------
__global__ void scatter_add(const float* __restrict__ feats,
                            const int* __restrict__ flats,
                            float* __restrict__ out) {
    int b  = blockIdx.y;
    int p0 = blockIdx.x * 32;
    int tx = threadIdx.x & 31;        // point within tile
    int ty = threadIdx.x >> 5;        // wave id in block -> channel group
    int flat = flats[b * NPTS + p0 + tx];
    if (flat >= RVOX) return;         // out-of-range segment is dropped
    const float* fp = feats + (size_t)b * CHANS * NPTS + p0 + tx;
    float*       op = out   + (size_t)b * CHANS * RVOX + flat;
#pragma unroll
    for (int i = 0; i < 8; ++i) {
        int c = ty * 8 + i;
        atomicAdd(op + (size_t)c * RVOX, fp[(size_t)c * NPTS]);
    }
}

// ---------------------------------------------------------------------------
// Phase 3: normalize via WMMA.  Per 16x16 tile:  D = S_tile x diag(1/max(cnt,1))
// using V_WMMA_F32_16X16X4_F32 (f32 exact), 4 K-slices of width 4.
// ISA 7.12.2 layouts:
//   A 16x4 f32 : lanes 0-15 -> M=lane, VGPR0=K0 VGPR1=K1; lanes 16-31 -> K2/K3
//   B 4x16 f32 : lanes 0-15 -> N=lane, VGPR0=k0 VGPR1=k1; lanes 16-31 -> k2/k3
//   C/D 16x16  : VGPR r -> row r (lanes 0-15), row r+8 (lanes 16-31)
// R and C are multiples of 16 -> full EXEC, no tail (WMMA requires EXEC all-1s).
// ---------------------------------------------------------------------------
__global__ void normalize_wmma(float* __restrict__ out,
                               const float* __restrict__ counts) {
    int wave = (blockIdx.x * blockDim.x + threadIdx.x) >> 5;
    int lane = threadIdx.x & 31;
    const int VT = RVOX / 16;                 // 1368 tiles along voxel dim
    int vt  = wave % VT;
    int rem = wave / VT;
    int ct  = rem & 3;                        // C/16 = 4
    int b   = rem >> 2;
    int v0  = vt * 16;
    int c0  = ct * 16;

    int n    = lane & 15;                     // = M for A, = N for B/D
    int half = lane >> 4;                     // lane half selects K pair / row+8
    int klo  = half * 2;

    float cnt = counts[b * RVOX + v0 + n];
    float inv = 1.0f / fmaxf(cnt, 1.0f);

    const float* abase =
        out + (size_t)b * CHANS * RVOX + (size_t)(c0 + n) * RVOX + v0;

    v8f acc = {0.f, 0.f, 0.f, 0.f, 0.f, 0.f, 0.f, 0.f};
#pragma unroll
    for (int g = 0; g < 4; ++g) {
        v2f a, bb;
        a.x  = abase[4 * g + klo];
        a.y  = abase[4 * g + klo + 1];
        bb.x = (n == 4 * g + klo)     ? inv : 0.0f;
        bb.y = (n == 4 * g + klo + 1) ? inv : 0.0f;
        // 8 args: (neg_a, A, neg_b, B, c_mod, C, reuse_a, reuse_b)
        acc = __builtin_amdgcn_wmma_f32_16x16x4_f32(
            false, a, false, bb, (short)0, acc, false, false);
    }

    float* obase =
        out + (size_t)b * CHANS * RVOX + (size_t)(c0 + 8 * half) * RVOX + v0 + n;
#pragma unroll
    for (int r = 0; r < 8; ++r) obase[(size_t)r * RVOX] = acc[r];
}

// ---------------------------------------------------------------------------
extern "C" void kernel_launch(void* const* d_in, const int* in_sizes, int n_in,
                              void* d_out, int out_size, void* d_ws, size_t ws_size,
                              hipStream_t stream) {
    (void)in_sizes; (void)n_in; (void)out_size; (void)ws_size;
    const float* feats  = (const float*)d_in[0];   // [B, C, N] f32
    const float* coords = (const float*)d_in[1];   // [B, N, 3] f32
    float* out = (float*)d_out;                    // [B, C, X, Y, Z] f32

    // workspace: counts [B*R] f32, then flats [B*N] i32  (~2.4 MB total)
    float* counts = (float*)d_ws;
    int*   flats  = (int*)((char*)d_ws + (size_t)BATCH * RVOX * sizeof(float));

    // Phase 0: zero output and counts
    int n4o = (BATCH * CHANS * RVOX) / 4;          // 5,603,328
    zero_f4<<<(n4o + 255) / 256, 256, 0, stream>>>((float4*)out, n4o);
    int n4c = (BATCH * RVOX) / 4;                  // 87,552
    zero_f4<<<(n4c + 255) / 256, 256, 0, stream>>>((float4*)counts, n4c);

    // Phase 1: voxel ids + counts
    flat_count<<<(BATCH * NPTS) / 256, 256, 0, stream>>>(coords, flats, counts);

    // Phase 2: scatter features
    dim3 gs(NPTS / 32, BATCH);
    scatter_add<<<gs, 256, 0, stream>>>(feats, flats, out);

    // Phase 3: WMMA diagonal-scale normalize (87552 tiles, 8 waves/block)
    int waves = BATCH * (CHANS / 16) * (RVOX / 16);
    normalize_wmma<<<waves / 8, 256, 0, stream>>>(out, counts);
}